// DeformableTransformerDecoderLayer_23974507446525
// MI455X (gfx1250) — compile-verified
//
#include <hip/hip_runtime.h>

typedef __attribute__((ext_vector_type(2))) float v2f;
typedef __attribute__((ext_vector_type(4))) float v4f;
typedef __attribute__((ext_vector_type(8))) float v8f;

#define B_   8
#define LQ_  300
#define C_   256
#define NH_  8
#define NL_  4
#define NP_  4
#define DFF_ 1024
#define HD_  32
#define S_   21760

// Try the gfx1250 async global->LDS copy path (ASYNCcnt tracked).
// If the assembler rejects the mnemonic, set to 0 (sync fallback below).
#define TRY_ASYNC_STAGE 1

__device__ __forceinline__ void wait_async0()
{
#if defined(__has_builtin) && __has_builtin(__builtin_amdgcn_s_wait_asynccnt)
  __builtin_amdgcn_s_wait_asynccnt(0);
#else
  asm volatile("s_wait_asynccnt 0x0" ::: "memory");
#endif
}

// ---------------------------------------------------------------------------
// Weight GEMM:  C[m,n] = relu?( sum_k A[m,k] * W[n,k] + bias[n] )
// A [M,K] row-major (lda=K), W [N,K] row-major, C ldc given.
// 128 threads = 4 waves; workgroup tile 64M x 64N; K chunked by KC=128 with
// the W slice staged in LDS (double buffered, async copy) and shared by all
// four waves. Each wave computes a 16M x 64N WMMA tile (4 x v8f accum).
// Requires K % 128 == 0 and N % 64 == 0 (true for every projection here).
// ---------------------------------------------------------------------------
#define KC 128
#define BROW (KC + 4)   // padded LDS row stride (floats): 528B, 16B aligned,
                        // rows land on distinct banks for fragment reads

__device__ __forceinline__ void stage_tile(float* __restrict__ dst,
                                           const float* __restrict__ W,
                                           int n0, int K, int kc0, int tid)
{
  // Copy 64 rows x KC floats of W (rows n0..n0+63, cols kc0..kc0+KC-1).
  // 2048 float4 transfers, 16 per thread.
#pragma unroll
  for (int i = 0; i < 16; ++i) {
    const int e4  = i * 128 + tid;   // float4 index within tile
    const int row = e4 >> 5;         // 32 float4 per row
    const int k4  = e4 & 31;
    const float* g = W + (long)(n0 + row) * K + kc0 + k4 * 4;
    float*       l = dst + row * BROW + k4 * 4;
#if TRY_ASYNC_STAGE
    const unsigned loff = (unsigned)(uintptr_t)l;   // low 32 bits = LDS offset
    asm volatile("global_load_async_to_lds_b128 %0, %1, off"
                 :: "v"(loff), "v"(g) : "memory");
#else
    *(v4f*)l = *(const v4f*)g;
#endif
  }
}

__global__ __launch_bounds__(128) void gemm_xwT_wmma(
    const float* __restrict__ A, const float* __restrict__ W,
    const float* __restrict__ bias, float* __restrict__ Cc,
    int M, int N, int K, int ldc, int relu)
{
  __shared__ float bs[2][64 * BROW];

  const int tid  = threadIdx.x;
  const int wave = tid >> 5;
  const int lane = tid & 31;
  const int half = lane >> 4;        // K-pair select (A/B f32 WMMA layout)
  const int l16  = lane & 15;

  const int m0 = blockIdx.x * 64 + wave * 16;
  const int n0 = blockIdx.y * 64;

  int arow = m0 + l16;
  if (arow >= M) arow = M - 1;       // clamp; stores masked later
  const float* arowp = A + (long)arow * K + half * 2;

  v8f acc0 = {}, acc1 = {}, acc2 = {}, acc3 = {};

  const int nchunks = K / KC;
  stage_tile(bs[0], W, n0, K, 0, tid);

  int cur = 0;
  for (int c = 0; c < nchunks; ++c) {
    wait_async0();
    __syncthreads();                 // bs[cur] holds chunk c for everyone
    if (c + 1 < nchunks)
      stage_tile(bs[cur ^ 1], W, n0, K, (c + 1) * KC, tid);

    const float* ap = arowp + c * KC;
    const float* b0 = &bs[cur][(l16 +  0) * BROW + half * 2];
    const float* b1 = &bs[cur][(l16 + 16) * BROW + half * 2];
    const float* b2 = &bs[cur][(l16 + 32) * BROW + half * 2];
    const float* b3 = &bs[cur][(l16 + 48) * BROW + half * 2];

#pragma unroll 8
    for (int kk = 0; kk < KC; kk += 4) {
      v2f af = *(const v2f*)(ap + kk);
      v2f f0 = *(const v2f*)(b0 + kk);
      v2f f1 = *(const v2f*)(b1 + kk);
      v2f f2 = *(const v2f*)(b2 + kk);
      v2f f3 = *(const v2f*)(b3 + kk);
      acc0 = __builtin_amdgcn_wmma_f32_16x16x4_f32(false, af, false, f0,
                                                   (short)0, acc0, false, false);
      acc1 = __builtin_amdgcn_wmma_f32_16x16x4_f32(false, af, false, f1,
                                                   (short)0, acc1, false, false);
      acc2 = __builtin_amdgcn_wmma_f32_16x16x4_f32(false, af, false, f2,
                                                   (short)0, acc2, false, false);
      acc3 = __builtin_amdgcn_wmma_f32_16x16x4_f32(false, af, false, f3,
                                                   (short)0, acc3, false, false);
    }
    __syncthreads();                 // all waves done reading bs[cur]
    cur ^= 1;
  }

  // Store: C/D layout -> VGPR j, lanes 0-15: M=j, N=lane; lanes 16-31: M=8+j.
  v8f accs[4] = {acc0, acc1, acc2, acc3};
#pragma unroll
  for (int t = 0; t < 4; ++t) {
    const int ncol = n0 + t * 16 + l16;          // N % 64 == 0: always valid
    const float bv = bias ? bias[ncol] : 0.0f;
#pragma unroll
    for (int j = 0; j < 8; ++j) {
      const int row = m0 + j + half * 8;
      if (row < M) {
        float v = accs[t][j] + bv;
        if (relu && v < 0.0f) v = 0.0f;
        Cc[(long)row * ldc + ncol] = v;
      }
    }
  }
}

// ---------------------------------------------------------------------------
// Generic strided/batched fp32 WMMA GEMM (used for the attention einsums).
//   C[m,n] = alpha * sum_k A[m,k] * B(k,n),  B(k,n) = Bp[k*bk + n*bn]
// One wave per 16M x 64N tile; blockIdx.z = b*nh + h with separate strides.
// ---------------------------------------------------------------------------
__global__ __launch_bounds__(32) void gemm_f32_wmma(
    const float* __restrict__ A, const float* __restrict__ Bp,
    const float* __restrict__ bias, float* __restrict__ Cc,
    int M, int N, int K,
    int lda, int bk, int bn, int ldc,
    int nh, long sAb, long sAh, long sBb, long sBh, long sCb, long sCh,
    float alpha, int relu)
{
  const int z = blockIdx.z;
  const int b = z / nh;
  const int h = z - b * nh;
  A  += (long)b * sAb + (long)h * sAh;
  Bp += (long)b * sBb + (long)h * sBh;
  Cc += (long)b * sCb + (long)h * sCh;

  const int m0   = blockIdx.x * 16;
  const int n0   = blockIdx.y * 64;
  const int lane = threadIdx.x;
  const int half = lane >> 4;
  const int l16  = lane & 15;

  int arow = m0 + l16;
  if (arow >= M) arow = M - 1;

  int ncol_c[4];
#pragma unroll
  for (int t = 0; t < 4; ++t) {
    int nc = n0 + t * 16 + l16;
    ncol_c[t] = (nc < N) ? nc : (N - 1);
  }

  v8f acc0 = {}, acc1 = {}, acc2 = {}, acc3 = {};

  const int kbase = half * 2;
  for (int k0 = 0; k0 < K; k0 += 4) {
    const int ka = k0 + kbase;
    v2f afrag = *(const v2f*)(A + (long)arow * lda + ka);

    v2f bf0, bf1, bf2, bf3;
    if (bk == 1) {
      bf0 = *(const v2f*)(Bp + (long)ncol_c[0] * bn + ka);
      bf1 = *(const v2f*)(Bp + (long)ncol_c[1] * bn + ka);
      bf2 = *(const v2f*)(Bp + (long)ncol_c[2] * bn + ka);
      bf3 = *(const v2f*)(Bp + (long)ncol_c[3] * bn + ka);
    } else {
      bf0.x = Bp[(long)ka * bk + (long)ncol_c[0] * bn];
      bf0.y = Bp[(long)(ka + 1) * bk + (long)ncol_c[0] * bn];
      bf1.x = Bp[(long)ka * bk + (long)ncol_c[1] * bn];
      bf1.y = Bp[(long)(ka + 1) * bk + (long)ncol_c[1] * bn];
      bf2.x = Bp[(long)ka * bk + (long)ncol_c[2] * bn];
      bf2.y = Bp[(long)(ka + 1) * bk + (long)ncol_c[2] * bn];
      bf3.x = Bp[(long)ka * bk + (long)ncol_c[3] * bn];
      bf3.y = Bp[(long)(ka + 1) * bk + (long)ncol_c[3] * bn];
    }

    acc0 = __builtin_amdgcn_wmma_f32_16x16x4_f32(false, afrag, false, bf0,
                                                 (short)0, acc0, false, false);
    acc1 = __builtin_amdgcn_wmma_f32_16x16x4_f32(false, afrag, false, bf1,
                                                 (short)0, acc1, false, false);
    acc2 = __builtin_amdgcn_wmma_f32_16x16x4_f32(false, afrag, false, bf2,
                                                 (short)0, acc2, false, false);
    acc3 = __builtin_amdgcn_wmma_f32_16x16x4_f32(false, afrag, false, bf3,
                                                 (short)0, acc3, false, false);
  }

  v8f accs[4] = {acc0, acc1, acc2, acc3};
#pragma unroll
  for (int t = 0; t < 4; ++t) {
    const int ncol = n0 + t * 16 + l16;
    if (ncol < N) {
      const float bv = bias ? bias[ncol] : 0.0f;
#pragma unroll
      for (int j = 0; j < 8; ++j) {
        const int row = m0 + j + half * 8;
        if (row < M) {
          float v = accs[t][j] * alpha + bv;
          if (relu && v < 0.0f) v = 0.0f;
          Cc[(long)row * ldc + ncol] = v;
        }
      }
    }
  }
}

// ---------------------------------------------------------------------------
__global__ __launch_bounds__(256) void add_kernel(const float* __restrict__ a,
                                                  const float* __restrict__ b,
                                                  float* __restrict__ o, int n)
{
  int i = blockIdx.x * 256 + threadIdx.x;
  if (i < n) o[i] = a[i] + b[i];
}

// Fused residual + LayerNorm over C=256.  One wave per row, 8 floats / lane.
__global__ __launch_bounds__(256) void ln_residual(
    const float* __restrict__ x, const float* __restrict__ y,
    const float* __restrict__ g, const float* __restrict__ bta,
    float* __restrict__ out, int rows)
{
  const int row = blockIdx.x * (blockDim.x >> 5) + (threadIdx.x >> 5);
  if (row >= rows) return;
  const int lane = threadIdx.x & 31;
  const float* xr = x + (long)row * C_;
  const float* yr = y + (long)row * C_;

  float v[8];
  float s = 0.0f;
#pragma unroll
  for (int i = 0; i < 8; ++i) {
    v[i] = xr[lane * 8 + i] + yr[lane * 8 + i];
    s += v[i];
  }
#pragma unroll
  for (int o = 16; o > 0; o >>= 1) s += __shfl_xor(s, o, 32);
  const float mean = s * (1.0f / C_);

  float var = 0.0f;
#pragma unroll
  for (int i = 0; i < 8; ++i) {
    const float d = v[i] - mean;
    var += d * d;
  }
#pragma unroll
  for (int o = 16; o > 0; o >>= 1) var += __shfl_xor(var, o, 32);
  const float inv = rsqrtf(var * (1.0f / C_) + 1e-5f);

  float* orow = out + (long)row * C_;
#pragma unroll
  for (int i = 0; i < 8; ++i) {
    const int c = lane * 8 + i;
    orow[c] = (v[i] - mean) * inv * g[c] + bta[c];
  }
}

// Row softmax (in place), one wave per row.
__global__ __launch_bounds__(256) void softmax_rows(float* __restrict__ p,
                                                    int rows, int n)
{
  const int row = blockIdx.x * (blockDim.x >> 5) + (threadIdx.x >> 5);
  if (row >= rows) return;
  const int lane = threadIdx.x & 31;
  float* pr = p + (long)row * n;

  float mx = -3.4e38f;
  for (int i = lane; i < n; i += 32) mx = fmaxf(mx, pr[i]);
#pragma unroll
  for (int o = 16; o > 0; o >>= 1) mx = fmaxf(mx, __shfl_xor(mx, o, 32));

  float s = 0.0f;
  for (int i = lane; i < n; i += 32) {
    const float e = expf(pr[i] - mx);
    pr[i] = e;
    s += e;
  }
#pragma unroll
  for (int o = 16; o > 0; o >>= 1) s += __shfl_xor(s, o, 32);
  const float inv = 1.0f / s;
  for (int i = lane; i < n; i += 32) pr[i] *= inv;
}

// Softmax over contiguous groups of 16 (deform attention weights), in place.
__global__ __launch_bounds__(256) void softmax16(float* __restrict__ p, int groups)
{
  const int g = blockIdx.x * 256 + threadIdx.x;
  if (g >= groups) return;
  float* pr = p + (long)g * 16;
  float mx = pr[0];
#pragma unroll
  for (int i = 1; i < 16; ++i) mx = fmaxf(mx, pr[i]);
  float s = 0.0f;
  float e[16];
#pragma unroll
  for (int i = 0; i < 16; ++i) { e[i] = expf(pr[i] - mx); s += e[i]; }
  const float inv = 1.0f / s;
#pragma unroll
  for (int i = 0; i < 16; ++i) pr[i] = e[i] * inv;
}

// ---------------------------------------------------------------------------
// Multi-scale deformable sampling. One wave per (b, lq, h); lane = channel d.
// ---------------------------------------------------------------------------
__device__ __forceinline__ float dtap(const float* __restrict__ value,
                                      long vbase, int st, int H, int W,
                                      int yi, int xi)
{
  const bool valid = (yi >= 0) & (yi < H) & (xi >= 0) & (xi < W);
  int yc = yi < 0 ? 0 : (yi > H - 1 ? H - 1 : yi);
  int xc = xi < 0 ? 0 : (xi > W - 1 ? W - 1 : xi);
  const float gv = value[vbase + (long)(st + yc * W + xc) * C_];
  return valid ? gv : 0.0f;
}

__global__ __launch_bounds__(128) void deform_sample(
    const float* __restrict__ value,   // [B,S,C]
    const float* __restrict__ refp,    // [B,LQ,NL,2]
    const float* __restrict__ offb,    // [B,LQ, NH*NL*NP*2]
    const float* __restrict__ awb,     // [B,LQ, NH*NL*NP] (softmaxed)
    float* __restrict__ samp)          // [B,LQ,C]
{
  const int wid = blockIdx.x * (blockDim.x >> 5) + (threadIdx.x >> 5);
  if (wid >= B_ * LQ_ * NH_) return;
  const int h  = wid % NH_;
  const int lq = (wid / NH_) % LQ_;
  const int b  = wid / (NH_ * LQ_);
  const int d  = threadIdx.x & 31;

  const int Hs[4] = {128, 64, 32, 16};
  const int Ws[4] = {128, 64, 32, 16};
  const int St[4] = {0, 16384, 20480, 21504};

  const float* rp = refp + ((long)b * LQ_ + lq) * NL_ * 2;
  const float* op = offb + ((long)b * LQ_ + lq) * (NH_ * NL_ * NP_ * 2) + h * NL_ * NP_ * 2;
  const float* ap = awb  + ((long)b * LQ_ + lq) * (NH_ * NL_ * NP_)     + h * NL_ * NP_;
  const long vbase = (long)b * S_ * C_ + h * HD_ + d;

  float acc = 0.0f;
#pragma unroll
  for (int lvl = 0; lvl < NL_; ++lvl) {
    const int H = Hs[lvl], W = Ws[lvl], st = St[lvl];
    const float rx = rp[lvl * 2 + 0];
    const float ry = rp[lvl * 2 + 1];
#pragma unroll
    for (int p = 0; p < NP_; ++p) {
      const float ox = op[(lvl * NP_ + p) * 2 + 0];
      const float oy = op[(lvl * NP_ + p) * 2 + 1];
      const float x = (rx + ox / (float)W) * (float)W - 0.5f;
      const float y = (ry + oy / (float)H) * (float)H - 0.5f;
      const float x0f = floorf(x), y0f = floorf(y);
      const float lx = x - x0f, ly = y - y0f;
      const int x0 = (int)x0f, y0 = (int)y0f;
      const float w00 = (1.0f - lx) * (1.0f - ly);
      const float w10 = lx * (1.0f - ly);
      const float w01 = (1.0f - lx) * ly;
      const float w11 = lx * ly;
      const float aw = ap[lvl * NP_ + p];
      float sampv = w00 * dtap(value, vbase, st, H, W, y0,     x0)
                  + w10 * dtap(value, vbase, st, H, W, y0,     x0 + 1)
                  + w01 * dtap(value, vbase, st, H, W, y0 + 1, x0)
                  + w11 * dtap(value, vbase, st, H, W, y0 + 1, x0 + 1);
      acc += aw * sampv;
    }
  }
  samp[((long)b * LQ_ + lq) * C_ + h * HD_ + d] = acc;
}

// ---------------------------------------------------------------------------
// Host-side orchestration
// ---------------------------------------------------------------------------
static inline void launch_gemm_w(hipStream_t s, const float* A, const float* W,
                                 const float* bias, float* Cc,
                                 int M, int N, int K, int ldc, int relu)
{
  dim3 grid((M + 63) / 64, N / 64);
  gemm_xwT_wmma<<<grid, 128, 0, s>>>(A, W, bias, Cc, M, N, K, ldc, relu);
}

static inline void launch_gemm(hipStream_t s,
    const float* A, const float* Bp, const float* bias, float* Cc,
    int M, int N, int K, int lda, int bk, int bn, int ldc,
    int batches, int nh, long sAb, long sAh, long sBb, long sBh,
    long sCb, long sCh, float alpha, int relu)
{
  dim3 grid((M + 15) / 16, (N + 63) / 64, batches);
  gemm_f32_wmma<<<grid, 32, 0, s>>>(A, Bp, bias, Cc, M, N, K, lda, bk, bn, ldc,
                                    nh, sAb, sAh, sBb, sBh, sCb, sCh, alpha, relu);
}

extern "C" void kernel_launch(void* const* d_in, const int* in_sizes, int n_in,
                              void* d_out, int out_size, void* d_ws, size_t ws_size,
                              hipStream_t stream)
{
  const float* tgt       = (const float*)d_in[0];
  const float* query_pos = (const float*)d_in[1];
  const float* ref_pts   = (const float*)d_in[2];
  const float* src       = (const float*)d_in[3];
  // d_in[4] = pos (unused); d_in[5], d_in[6] int shapes (hardcoded)
  const float* self_in_w  = (const float*)d_in[7];
  const float* self_in_b  = (const float*)d_in[8];
  const float* self_out_w = (const float*)d_in[9];
  const float* self_out_b = (const float*)d_in[10];
  const float* norm2_g    = (const float*)d_in[11];
  const float* norm2_b    = (const float*)d_in[12];
  const float* off_w      = (const float*)d_in[13];
  const float* off_b      = (const float*)d_in[14];
  const float* aw_w       = (const float*)d_in[15];
  const float* aw_b       = (const float*)d_in[16];
  const float* val_w      = (const float*)d_in[17];
  const float* val_b      = (const float*)d_in[18];
  const float* outp_w     = (const float*)d_in[19];
  const float* outp_b     = (const float*)d_in[20];
  const float* norm1_g    = (const float*)d_in[21];
  const float* norm1_b    = (const float*)d_in[22];
  const float* lin1_w     = (const float*)d_in[23];
  const float* lin1_b     = (const float*)d_in[24];
  const float* lin2_w     = (const float*)d_in[25];
  const float* lin2_b     = (const float*)d_in[26];
  const float* norm3_g    = (const float*)d_in[27];
  const float* norm3_b    = (const float*)d_in[28];
  float* out = (float*)d_out;
  float* ws  = (float*)d_ws;

  const int M  = B_ * LQ_;           // 2400
  const int MS = B_ * S_;            // 174080
  const int NT = M * C_;             // 614400

  // Workspace layout (floats), lifetime-based reuse.
  float* qin    = ws + 0;            // [M, C]
  float* qk     = ws + 614400;       // [M, 2C]   Q | K
  float* vbuf   = ws + 1843200;      // [M, C]
  float* scores = ws + 2457600;      // [B, NH, LQ, LQ]
  float* obuf   = ws + 8217600;      // [M, C]
  float* mo     = ws + 8832000;      // [M, C]
  float* tbuf   = ws + 9446400;      // [M, C]
  float* t1buf  = ws + 10060800;     // [M, C]
  float* valb   = ws + 10675200;     // [B, S, C]
  // Reuse of [0 .. 8832000) after tbuf is computed:
  float* tq     = ws + 0;            // [M, C]
  float* offb2  = ws + 614400;       // [M, 256]
  float* awb2   = ws + 1228800;      // [M, 128]
  float* sampb  = ws + 1536000;      // [M, C]
  float* t2b    = ws + 2150400;      // [M, C]
  float* h1b    = ws + 2764800;      // [M, DFF]
  float* ffnb   = ws + 5222400;      // [M, C]

  const float inv_sqrt_hd = 0.17677669529663687f; // 1/sqrt(32)

  // ---- Self attention -----------------------------------------------------
  add_kernel<<<(NT + 255) / 256, 256, 0, stream>>>(tgt, query_pos, qin, NT);
  // Q,K = q_in @ in_w[:2C].T + in_b[:2C]
  launch_gemm_w(stream, qin, self_in_w, self_in_b, qk, M, 2 * C_, C_, 2 * C_, 0);
  // V = tgt @ in_w[2C:].T + in_b[2C:]
  launch_gemm_w(stream, tgt, self_in_w + (long)2 * C_ * C_, self_in_b + 2 * C_,
                vbuf, M, C_, C_, C_, 0);
  // scores[b,h] = Q_bh @ K_bh^T / sqrt(HD)
  launch_gemm(stream, qk, qk + C_, nullptr, scores,
              LQ_, LQ_, HD_, 2 * C_, 1, 2 * C_, LQ_,
              B_ * NH_, NH_,
              (long)LQ_ * 2 * C_, HD_,
              (long)LQ_ * 2 * C_, HD_,
              (long)NH_ * LQ_ * LQ_, (long)LQ_ * LQ_,
              inv_sqrt_hd, 0);
  softmax_rows<<<(B_ * NH_ * LQ_ + 7) / 8, 256, 0, stream>>>(scores, B_ * NH_ * LQ_, LQ_);
  // o[b,h] = att @ V_bh
  launch_gemm(stream, scores, vbuf, nullptr, obuf,
              LQ_, HD_, LQ_, LQ_, C_, 1, C_,
              B_ * NH_, NH_,
              (long)NH_ * LQ_ * LQ_, (long)LQ_ * LQ_,
              (long)LQ_ * C_, HD_,
              (long)LQ_ * C_, HD_,
              1.0f, 0);
  // mha_out = o @ out_w.T + out_b
  launch_gemm_w(stream, obuf, self_out_w, self_out_b, mo, M, C_, C_, C_, 0);
  // t = LN(tgt + mha_out)
  ln_residual<<<(M + 7) / 8, 256, 0, stream>>>(tgt, mo, norm2_g, norm2_b, tbuf, M);

  // ---- Deformable attention ----------------------------------------------
  add_kernel<<<(NT + 255) / 256, 256, 0, stream>>>(tbuf, query_pos, tq, NT);
  launch_gemm_w(stream, tq, off_w, off_b, offb2, M, NH_ * NL_ * NP_ * 2, C_,
                NH_ * NL_ * NP_ * 2, 0);
  launch_gemm_w(stream, tq, aw_w, aw_b, awb2, M, NH_ * NL_ * NP_, C_,
                NH_ * NL_ * NP_, 0);
  softmax16<<<(B_ * LQ_ * NH_ + 255) / 256, 256, 0, stream>>>(awb2, B_ * LQ_ * NH_);
  // value = src @ val_w.T + val_b   (174080 x 256 x 256)
  launch_gemm_w(stream, src, val_w, val_b, valb, MS, C_, C_, C_, 0);
  deform_sample<<<(B_ * LQ_ * NH_ + 3) / 4, 128, 0, stream>>>(valb, ref_pts, offb2, awb2, sampb);
  // t2 = samp @ outp_w.T + outp_b
  launch_gemm_w(stream, sampb, outp_w, outp_b, t2b, M, C_, C_, C_, 0);
  // t1 = LN(t + t2)
  ln_residual<<<(M + 7) / 8, 256, 0, stream>>>(tbuf, t2b, norm1_g, norm1_b, t1buf, M);

  // ---- FFN ----------------------------------------------------------------
  launch_gemm_w(stream, t1buf, lin1_w, lin1_b, h1b, M, DFF_, C_, DFF_, 1);
  launch_gemm_w(stream, h1b, lin2_w, lin2_b, ffnb, M, C_, DFF_, C_, 0);
  ln_residual<<<(M + 7) / 8, 256, 0, stream>>>(t1buf, ffnb, norm3_g, norm3_b, out, M);

  (void)in_sizes; (void)n_in; (void)out_size; (void)ws_size;
}